// HOGExtractor_14370960572648
// MI455X (gfx1250) — compile-verified
//
#include <hip/hip_runtime.h>
#include <hip/hip_bf16.h>
#include <math.h>

// HOG extractor for MI455X (gfx1250). Memory-bound (~235 MB traffic -> ~10us
// floor at 23.3 TB/s). No matmul structure -> WMMA inapplicable; instead use
// the CDNA5 async global->LDS path (GLOBAL_LOAD_ASYNC_TO_LDS_B128 + ASYNCcnt)
// with double-buffered band staging.

#define HH 256
#define WW 128
#define NBINS 9
#define CELLSZ 16
#define CYC 16            // cells in y
#define CXC 8             // cells in x
#define BYB 15            // blocks in y
#define BXB 7             // blocks in x
#define FEATN 3780
#define PADTO 4096
#define BAND_ROWS 18                      // 16 + 2 halo rows
#define BAND_F (3 * BAND_ROWS * WW)       // 6912 floats per band
#define BAND_F4 (BAND_F / 4)              // 1728 float4 chunks
#define NTHREADS 256

// Issue one band's async global->LDS loads (B128 per lane, GVS addressing:
// SGPR64 base + per-lane signed 32-bit byte offset). EXEC-masked tail.
__device__ __forceinline__ void issue_band(const float* __restrict__ base,
                                           int bandIdx, unsigned ldsBase, int tid) {
  const int y0 = bandIdx * CELLSZ;
#pragma unroll
  for (int it = 0; it < 7; ++it) {
    int f = it * NTHREADS + tid;
    if (f < BAND_F4) {
      int row_ch = f >> 5;                 // 0..53 : ch*18 + r
      int col4   = f & 31;                 // float4 column
      int ch = row_ch / BAND_ROWS;
      int rr = row_ch - ch * BAND_ROWS;    // 0..17
      int y  = y0 - 1 + rr;
      y = y < 0 ? 0 : (y > HH - 1 ? HH - 1 : y);   // edge replication
      int goff = ((ch << 15) + (y << 7) + (col4 << 2)) << 2;            // bytes
      unsigned loff = ldsBase + (unsigned)(((row_ch << 7) + (col4 << 2)) << 2);
      asm volatile("global_load_async_to_lds_b128 %0, %1, %2"
                   :: "v"(loff), "v"(goff), "s"(base)
                   : "memory");
    }
  }
}

__device__ __forceinline__ void compute_band(const float* __restrict__ buf,
                                             float* __restrict__ hist,
                                             int bandIdx, int tid) {
  const int ly  = tid >> 4;          // 0..15 : row inside the band's cells
  const int l16 = tid & 15;
  const int x0  = l16 << 3;          // 8 consecutive pixels, within one cell
  const int cx  = l16 >> 1;          // cell column 0..7
  const int r   = ly + 1;            // LDS row (halo at 0 and 17)
  float* hc = &hist[((bandIdx << 3) + cx) * NBINS];
  const float PI_F = 3.14159265358979f;
#pragma unroll
  for (int k = 0; k < 8; ++k) {
    int x  = x0 + k;
    int xm = (x == 0) ? 0 : x - 1;
    int xp = (x == WW - 1) ? WW - 1 : x + 1;
    float bestm = -1.0f, gdx = 0.0f, gdy = 0.0f;
#pragma unroll
    for (int ch = 0; ch < 3; ++ch) {
      const float* Lr = buf + (ch * BAND_ROWS + r) * WW;
      float dxc = Lr[xp] - Lr[xm];
      float dyc = Lr[WW + x] - Lr[x - WW];
      float m = dxc * dxc + dyc * dyc;
      if (m > bestm) { bestm = m; gdx = dxc; gdy = dyc; }   // first-max wins
    }
    float mag = sqrtf(gdx * gdx + gdy * gdy);
    float a = atan2f(gdy, gdx);          // (-pi, pi]
    if (a < 0.0f) a += PI_F;             // mod pi -> [0, pi)
    if (a >= PI_F) a -= PI_F;
    float pos  = a * (9.0f / PI_F) - 0.5f;
    float b0f  = floorf(pos);
    float frac = pos - b0f;
    int b0  = (int)b0f;                                  // -1..8
    int b0i = b0 < 0 ? (NBINS - 1) : (b0 >= NBINS ? 0 : b0);
    int b1i = (b0i + 1 == NBINS) ? 0 : b0i + 1;
    atomicAdd(&hc[b0i], mag * (1.0f - frac));
    atomicAdd(&hc[b1i], mag * frac);
  }
}

__global__ __launch_bounds__(NTHREADS)
void hog_kernel(const float* __restrict__ img, float* __restrict__ out) {
  __shared__ float band[2][BAND_F];               // 2 x 27648 B double buffer
  __shared__ float hist[CYC * CXC * NBINS];       // 1152 floats

  const int b   = blockIdx.x;
  const int tid = threadIdx.x;
  const float* base = img + (size_t)b * (3 * HH * WW);

  for (int idx = tid; idx < CYC * CXC * NBINS; idx += NTHREADS) hist[idx] = 0.0f;

  unsigned lds0 = (unsigned)(size_t)&band[0][0];  // low 32 bits == LDS offset
  unsigned lds1 = (unsigned)(size_t)&band[1][0];

  // Prologue: stage band 0.
  issue_band(base, 0, lds0, tid);
  asm volatile("s_wait_asynccnt 0" ::: "memory");
  __syncthreads();

  for (int i = 0; i < CYC; ++i) {
    const float* curPtr = band[i & 1];
    unsigned nxtLds = (i & 1) ? lds0 : lds1;
    if (i + 1 < CYC) issue_band(base, i + 1, nxtLds, tid);   // overlap with compute
    compute_band(curPtr, hist, i, tid);
    asm volatile("s_wait_asynccnt 0" ::: "memory");          // band i+1 landed
    __syncthreads();                                         // all waves done; buffers swap
  }

  // Block normalization: 105 blocks of 36 values, one block per thread.
  float* outb = out + (size_t)b * PADTO;
  if (tid < BYB * BXB) {
    int by = tid / BXB;
    int bx = tid - by * BXB;
    const float* c00 = &hist[((by << 3) + bx) * NBINS];
    float v[36];
#pragma unroll
    for (int j = 0; j < NBINS; ++j) {
      v[j]      = c00[j];
      v[9 + j]  = c00[NBINS + j];                 // cell (by, bx+1)
      v[18 + j] = c00[CXC * NBINS + j];           // cell (by+1, bx)
      v[27 + j] = c00[CXC * NBINS + NBINS + j];   // cell (by+1, bx+1)
    }
    float ss = 0.0f;
#pragma unroll
    for (int j = 0; j < 36; ++j) ss += v[j] * v[j];
    float s1 = 1.0f / (sqrtf(ss) + 36.0f * 0.1f);
    float ss2 = 0.0f;
#pragma unroll
    for (int j = 0; j < 36; ++j) {
      float t = v[j] * s1;
      t = t > 0.2f ? 0.2f : t;                    // values are non-negative
      v[j] = t;
      ss2 += t * t;
    }
    float s2 = 1.0f / (sqrtf(ss2) + 1e-3f);
    float* o = outb + tid * 36;
#pragma unroll
    for (int j = 0; j < 36; ++j) o[j] = v[j] * s2;
  }
  // Zero the pad 3780..4095.
  for (int idx = FEATN + tid; idx < PADTO; idx += NTHREADS) outb[idx] = 0.0f;
}

extern "C" void kernel_launch(void* const* d_in, const int* in_sizes, int n_in,
                              void* d_out, int out_size, void* d_ws, size_t ws_size,
                              hipStream_t stream) {
  (void)d_ws; (void)ws_size; (void)out_size;
  const float* images = (const float*)d_in[0];
  int B = in_sizes[0] / (3 * HH * WW);
  if (B <= 0) return;

  hog_kernel<<<B, NTHREADS, 0, stream>>>(images, (float*)d_out);

  // Append targets (i32) and camid (i32) after the [B,4096] f32 features.
  size_t featBytes = (size_t)B * PADTO * sizeof(float);
  if (n_in >= 3) {
    hipMemcpyAsync((char*)d_out + featBytes, d_in[1],
                   (size_t)in_sizes[1] * sizeof(int), hipMemcpyDeviceToDevice, stream);
    hipMemcpyAsync((char*)d_out + featBytes + (size_t)in_sizes[1] * sizeof(int), d_in[2],
                   (size_t)in_sizes[2] * sizeof(int), hipMemcpyDeviceToDevice, stream);
  }
}